// NMRGraphEncoder_48704929136702
// MI455X (gfx1250) — compile-verified
//
#include <hip/hip_runtime.h>

typedef __attribute__((ext_vector_type(2))) float v2f;
typedef __attribute__((ext_vector_type(8))) float v8f;

// ---------------------------------------------------------------------------
// GEMM: out[N,64] = act(in[N,64]) @ W[64,64] + b   for 4 weight sets at once.
// Block = 4 wave32s; wave w owns weight matrix w (uniform per wave -> EXEC all
// ones for WMMA). Each wave computes the full 16x64 output stripe of its
// matrix: 4 accumulators, 16 K-steps, 64 chained V_WMMA_F32_16X16X4_F32.
// A fragment is loaded once per K-step and reused by 4 independent WMMAs.
//
// A frag (16x4 f32):  lanes 0-15 hold row M=lane, K={0,1}; lanes 16-31 K={2,3}
// B frag (4x16 f32):  lanes 0-15 hold col N=lane, K={0,1}; lanes 16-31 K={2,3}
// C/D (16x16 f32):    VGPR j = row j (lanes 0-15) / row j+8 (lanes 16-31)
// ---------------------------------------------------------------------------
template <bool RELU>
__global__ __launch_bounds__(128) void gemm4_wmma(
    const float* __restrict__ x, int N,
    const float* __restrict__ W0, const float* __restrict__ B0, float* __restrict__ O0,
    const float* __restrict__ W1, const float* __restrict__ B1, float* __restrict__ O1,
    const float* __restrict__ W2, const float* __restrict__ B2, float* __restrict__ O2,
    const float* __restrict__ W3, const float* __restrict__ B3, float* __restrict__ O3)
{
    const int wv = threadIdx.x >> 5;         // wave id -> which weight matrix
    const float* W; const float* B; float* O;
    switch (wv) {
        case 0:  W = W0; B = B0; O = O0; break;
        case 1:  W = W1; B = B1; O = O1; break;
        case 2:  W = W2; B = B2; O = O2; break;
        default: W = W3; B = B3; O = O3; break;
    }
    const int lane = threadIdx.x & 31;
    const int r0   = blockIdx.x * 16;        // row tile base
    const int lr   = lane & 15;
    const int khi  = lane >> 4;              // 0: K lanes {0,1}, 1: K lanes {2,3}

    // C starts as the bias (broadcast along rows)
    v8f acc[4];
#pragma unroll
    for (int c = 0; c < 4; ++c) {
        float bias = B[c * 16 + lr];
#pragma unroll
        for (int j = 0; j < 8; ++j) acc[c][j] = bias;
    }

    int arow = r0 + lr; if (arow >= N) arow = N - 1;   // clamp reads, EXEC stays full
    const float* xr = x + (size_t)arow * 64 + khi * 2;

#pragma unroll
    for (int kk = 0; kk < 16; ++kk) {
        v2f a;
        float a0 = xr[kk * 4 + 0];
        float a1 = xr[kk * 4 + 1];
        if (RELU) { a0 = fmaxf(a0, 0.f); a1 = fmaxf(a1, 0.f); }
        a[0] = a0; a[1] = a1;
        const int kb = kk * 4 + khi * 2;
        const float* wr0 = W + (size_t)(kb + 0) * 64 + lr;
        const float* wr1 = W + (size_t)(kb + 1) * 64 + lr;
#pragma unroll
        for (int c = 0; c < 4; ++c) {
            v2f b;
            b[0] = wr0[c * 16];
            b[1] = wr1[c * 16];
            acc[c] = __builtin_amdgcn_wmma_f32_16x16x4_f32(
                /*neg_a=*/false, a, /*neg_b=*/false, b,
                /*c_mod=*/(short)0, acc[c], /*reuse_a=*/false, /*reuse_b=*/false);
        }
    }

#pragma unroll
    for (int c = 0; c < 4; ++c)
#pragma unroll
        for (int j = 0; j < 8; ++j) {
            int row = r0 + j + khi * 8;
            if (row < N) O[(size_t)row * 64 + c * 16 + lr] = acc[c][j];
        }
}

// ---------------------------------------------------------------------------
// Edge-phase kernels (one thread per (edge, head)); all L2-resident gathers.
// ---------------------------------------------------------------------------
__device__ __forceinline__ void atomicMaxF(float* addr, float val) {
    if (val >= 0.f) atomicMax((int*)addr, __float_as_int(val));
    else            atomicMin((unsigned int*)addr, (unsigned int)__float_as_int(val));
}

__global__ void init_softmax(float* __restrict__ smax, float* __restrict__ den, int n4) {
    int t = blockIdx.x * blockDim.x + threadIdx.x;
    if (t < n4) { smax[t] = -__builtin_inff(); den[t] = 0.f; }
}

__global__ void edge_score(const float* __restrict__ q, const float* __restrict__ k,
                           const float* __restrict__ ea, const int* __restrict__ srcI,
                           const int* __restrict__ dstI, const float* __restrict__ We,
                           float* __restrict__ score, float* __restrict__ smax, int E)
{
    int t = blockIdx.x * blockDim.x + threadIdx.x;
    if (t >= E * 4) return;
    int e = t >> 2, h = t & 3;
    int s = srcI[e], d = dstI[e];
    float a = ea[e];
    const float4* qp = (const float4*)(q + (size_t)d * 64 + h * 16);
    const float4* kp = (const float4*)(k + (size_t)s * 64 + h * 16);
    const float4* wp = (const float4*)(We + h * 16);
    float acc = 0.f;
#pragma unroll
    for (int i = 0; i < 4; ++i) {
        float4 qq = qp[i], kk = kp[i], ww = wp[i];
        acc += qq.x * (kk.x + a * ww.x) + qq.y * (kk.y + a * ww.y)
             + qq.z * (kk.z + a * ww.z) + qq.w * (kk.w + a * ww.w);
    }
    acc *= 0.25f;                       // 1/sqrt(C), C=16
    score[t] = acc;
    atomicMaxF(&smax[d * 4 + h], acc);
}

__global__ void edge_exp(const int* __restrict__ dstI, float* __restrict__ score,
                         const float* __restrict__ smax, float* __restrict__ den, int E)
{
    int t = blockIdx.x * blockDim.x + threadIdx.x;
    if (t >= E * 4) return;
    int e = t >> 2, h = t & 3;
    int d = dstI[e];
    float ex = __expf(score[t] - smax[d * 4 + h]);
    score[t] = ex;                      // in-place: score buffer now holds exp values
    atomicAdd(&den[d * 4 + h], ex);
}

__global__ void edge_scatter(const float* __restrict__ v, const float* __restrict__ ea,
                             const int* __restrict__ srcI, const int* __restrict__ dstI,
                             const float* __restrict__ We, const float* __restrict__ exb,
                             const float* __restrict__ den, float* __restrict__ out, int E)
{
    int t = blockIdx.x * blockDim.x + threadIdx.x;
    if (t >= E * 4) return;
    int e = t >> 2, h = t & 3;
    int s = srcI[e], d = dstI[e];
    float a = ea[e];
    float alpha = exb[t] / (den[d * 4 + h] + 1e-16f);
    const float4* vp = (const float4*)(v + (size_t)s * 64 + h * 16);
    const float4* wp = (const float4*)(We + h * 16);
    float* op = out + (size_t)d * 64 + h * 16;
#pragma unroll
    for (int i = 0; i < 4; ++i) {
        float4 vv = vp[i], ww = wp[i];
        atomicAdd(op + i * 4 + 0, alpha * (vv.x + a * ww.x));
        atomicAdd(op + i * 4 + 1, alpha * (vv.y + a * ww.y));
        atomicAdd(op + i * 4 + 2, alpha * (vv.z + a * ww.z));
        atomicAdd(op + i * 4 + 3, alpha * (vv.w + a * ww.w));
    }
}

// ---------------------------------------------------------------------------
// Pool + graph MLP tail
// ---------------------------------------------------------------------------
__global__ void zero_f32(float* __restrict__ p, int n) {
    int t = blockIdx.x * blockDim.x + threadIdx.x;
    if (t < n) p[t] = 0.f;
}

__global__ void pool_sum(const float* __restrict__ h, const int* __restrict__ batch,
                         float* __restrict__ sums, float* __restrict__ cnt, int N)
{
    int t = blockIdx.x * blockDim.x + threadIdx.x;
    if (t >= N * 64) return;
    int n = t >> 6, d = t & 63;
    int g = batch[n];
    atomicAdd(&sums[g * 64 + d], h[t]);
    if (d == 0) atomicAdd(&cnt[g], 1.0f);
}

__global__ __launch_bounds__(128) void graph_mlp(
    const float* __restrict__ sums, const float* __restrict__ cnt,
    const float* __restrict__ W1, const float* __restrict__ B1,
    const float* __restrict__ W2, const float* __restrict__ B2,
    const float* __restrict__ W3, const float* __restrict__ B3,
    float* __restrict__ out)
{
    __shared__ float g[64];
    __shared__ float h1[64];
    __shared__ float h2[16];
    const int gi = blockIdx.x;
    const int t  = threadIdx.x;
    float c = fmaxf(cnt[gi], 1.0f);
    if (t < 64) g[t] = sums[gi * 64 + t] / c;
    __syncthreads();
    if (t < 64) {
        float acc = B1[t];
        for (int i = 0; i < 64; ++i) acc += g[i] * W1[i * 64 + t];
        h1[t] = fmaxf(acc, 0.f);
    }
    __syncthreads();
    if (t < 16) {
        float acc = B2[t];
        for (int i = 0; i < 64; ++i) acc += h1[i] * W2[i * 16 + t];
        h2[t] = fmaxf(acc, 0.f);
    }
    __syncthreads();
    {
        float acc = B3[t];
        for (int i = 0; i < 16; ++i) acc += h2[i] * W3[i * 128 + t];
        out[(size_t)gi * 128 + t] = acc;
    }
}

// ---------------------------------------------------------------------------
extern "C" void kernel_launch(void* const* d_in, const int* in_sizes, int n_in,
                              void* d_out, int out_size, void* d_ws, size_t ws_size,
                              hipStream_t stream)
{
    const float* x     = (const float*)d_in[0];
    const float* ea    = (const float*)d_in[1];
    const int*   ei    = (const int*)  d_in[2];
    const int*   batch = (const int*)  d_in[3];
    const float* Wq = (const float*)d_in[4];
    const float* bq = (const float*)d_in[5];
    const float* Wk = (const float*)d_in[6];
    const float* bk = (const float*)d_in[7];
    const float* Wv = (const float*)d_in[8];
    const float* bv = (const float*)d_in[9];
    const float* We = (const float*)d_in[10];
    const float* Ws = (const float*)d_in[11];
    const float* bs = (const float*)d_in[12];
    const float* W1 = (const float*)d_in[13];
    const float* b1 = (const float*)d_in[14];
    const float* W2 = (const float*)d_in[15];
    const float* b2 = (const float*)d_in[16];
    const float* W3 = (const float*)d_in[17];
    const float* b3 = (const float*)d_in[18];

    const int N = in_sizes[0] / 64;
    const int E = in_sizes[1];
    const int G = (out_size - N * 64) / 128;

    // workspace carve-up (floats)
    float* ws   = (float*)d_ws;
    float* q    = ws; ws += (size_t)N * 64;
    float* k    = ws; ws += (size_t)N * 64;
    float* v    = ws; ws += (size_t)N * 64;
    float* hA   = ws; ws += (size_t)N * 64;
    float* hB   = ws; ws += (size_t)N * 64;
    float* sc   = ws; ws += (size_t)E * 4;   // score, then exp (in place)
    float* smax = ws; ws += (size_t)N * 4;
    float* den  = ws; ws += (size_t)N * 4;
    float* sums = ws; ws += (size_t)G * 64;
    float* cnt  = ws; ws += (size_t)G;       // contiguous with sums

    const int* srcI = ei;
    const int* dstI = ei + E;

    float* nodeOut  = (float*)d_out;
    float* graphOut = nodeOut + (size_t)N * 64;

    const int n4 = N * 4;
    const int e4 = E * 4;
    const int gemmBlocks = (N + 15) / 16;

    const float* hin = x;
    for (int l = 0; l < 3; ++l) {
        float* hout = (l == 0) ? hA : (l == 1) ? hB : nodeOut;
        // q/k/v GEMMs + skip GEMM (skip result seeds the scatter accumulator)
        if (l == 0) {
            gemm4_wmma<false><<<gemmBlocks, 128, 0, stream>>>(hin, N,
                Wq + (size_t)l * 64 * 64, bq + l * 64, q,
                Wk + (size_t)l * 64 * 64, bk + l * 64, k,
                Wv + (size_t)l * 64 * 64, bv + l * 64, v,
                Ws + (size_t)l * 64 * 64, bs + l * 64, hout);
        } else {
            gemm4_wmma<true><<<gemmBlocks, 128, 0, stream>>>(hin, N,
                Wq + (size_t)l * 64 * 64, bq + l * 64, q,
                Wk + (size_t)l * 64 * 64, bk + l * 64, k,
                Wv + (size_t)l * 64 * 64, bv + l * 64, v,
                Ws + (size_t)l * 64 * 64, bs + l * 64, hout);
        }
        init_softmax<<<(n4 + 255) / 256, 256, 0, stream>>>(smax, den, n4);
        edge_score  <<<(e4 + 255) / 256, 256, 0, stream>>>(q, k, ea, srcI, dstI,
                                                           We + l * 64, sc, smax, E);
        edge_exp    <<<(e4 + 255) / 256, 256, 0, stream>>>(dstI, sc, smax, den, E);
        edge_scatter<<<(e4 + 255) / 256, 256, 0, stream>>>(v, ea, srcI, dstI,
                                                           We + l * 64, sc, den, hout, E);
        hin = hout;
    }

    // global mean pool + graph MLP
    const int pz = G * 64 + G;
    zero_f32 <<<(pz + 255) / 256, 256, 0, stream>>>(sums, pz);
    pool_sum <<<(N * 64 + 255) / 256, 256, 0, stream>>>(nodeOut, batch, sums, cnt, N);
    graph_mlp<<<G, 128, 0, stream>>>(sums, cnt, W1, b1, W2, b2, W3, b3, graphOut);
}